// GATModel_32976758899208
// MI455X (gfx1250) — compile-verified
//
#include <hip/hip_runtime.h>
#include <hip/hip_bf16.h>
#include <math.h>

// ---------------------------------------------------------------------------
// GAT (3 layers, heads=4/4/1) for MI455X (gfx1250, wave32).
// GEMMs use V_WMMA_F32_16X16X4_F32 (full fp32 precision on the matrix pipe).
// Edge phase is L2-resident (total working set ~70MB << 192MB L2).
// ---------------------------------------------------------------------------

typedef __attribute__((ext_vector_type(2))) float v2f;
typedef __attribute__((ext_vector_type(8))) float v8f;

#define NEG_SLOPE 0.2f

// Monotone float <-> uint mapping so atomicMax(u32) implements float max.
__device__ __forceinline__ unsigned enc_ord(float f) {
  unsigned u = __float_as_uint(f);
  return (u & 0x80000000u) ? ~u : (u | 0x80000000u);
}
__device__ __forceinline__ float dec_ord(unsigned u) {
  unsigned b = (u & 0x80000000u) ? (u & 0x7FFFFFFFu) : ~u;
  return __uint_as_float(b);
}
#define ENC_NEG_INF 0x007FFFFFu  // enc_ord(-inf)

// ---------------------------------------------------------------------------
// 0) Per-layer init: zero accumulators / denominators, seed max with -inf.
// ---------------------------------------------------------------------------
__global__ void init_layer_kernel(float* __restrict__ acc, int nm,
                                  float* __restrict__ den,
                                  unsigned* __restrict__ mlog, int nh) {
  int i = blockIdx.x * blockDim.x + threadIdx.x;
  if (i < nm) acc[i] = 0.0f;
  if (i < nh) { den[i] = 0.0f; mlog[i] = ENC_NEG_INF; }
}

// ---------------------------------------------------------------------------
// 1) Hout[N,M] = X[N,K] @ W[K,M] via V_WMMA_F32_16X16X4_F32.
//    One wave -> 16x64 output tile (4 accumulators, A fragment reused 4x).
//    A layout (32-bit 16x4): lanes 0-15 hold K=k+0 (v.x), k+1 (v.y);
//                            lanes 16-31 hold K=k+2, k+3.
//    B layout (4x16): v.x row k+kh, v.y row k+kh+1, column = lane&15.
//    C/D layout: VGPR v -> row v (lanes 0-15) / v+8 (lanes 16-31), col lane&15.
//    Requires N % 16 == 0 (N=10000 = 625*16) and K % 4 == 0.
// ---------------------------------------------------------------------------
__global__ __launch_bounds__(128)
void gemm_wmma_f32(const float* __restrict__ X, const float* __restrict__ W,
                   float* __restrict__ Hout, int K, int M) {
  const int wave = threadIdx.x >> 5;
  const int lane = threadIdx.x & 31;
  const int lr   = lane & 15;          // row-in-tile (A) / col-in-tile (B)
  const int kh   = (lane >> 4) << 1;   // 0 for lanes 0-15, 2 for lanes 16-31

  const int rowBase = blockIdx.x << 4;
  const int nBase   = (blockIdx.y * 4 + wave) << 6;  // 64 cols per wave
  if (nBase >= M) return;

  v8f c0 = {}, c1 = {}, c2 = {}, c3 = {};
  const float* __restrict__ arow = X + (size_t)(rowBase + lr) * K;

  for (int k = 0; k < K; k += 4) {
    v2f a;
    a.x = arow[k + kh];
    a.y = arow[k + kh + 1];

    const float* __restrict__ wp = W + (size_t)(k + kh) * M + nBase + lr;
    v2f b0, b1, b2, b3;
    b0.x = wp[0];   b0.y = wp[M];
    b1.x = wp[16];  b1.y = wp[M + 16];
    b2.x = wp[32];  b2.y = wp[M + 32];
    b3.x = wp[48];  b3.y = wp[M + 48];

    c0 = __builtin_amdgcn_wmma_f32_16x16x4_f32(false, a, false, b0, (short)0, c0, false, false);
    c1 = __builtin_amdgcn_wmma_f32_16x16x4_f32(false, a, false, b1, (short)0, c1, false, false);
    c2 = __builtin_amdgcn_wmma_f32_16x16x4_f32(false, a, false, b2, (short)0, c2, false, false);
    c3 = __builtin_amdgcn_wmma_f32_16x16x4_f32(false, a, false, b3, (short)0, c3, false, false);
  }

  const int colOff = lane & 15;
  const int rowOff = (lane >> 4) << 3;
#pragma unroll
  for (int v = 0; v < 8; ++v) {
    float* __restrict__ op =
        Hout + (size_t)(rowBase + rowOff + v) * M + nBase + colOff;
    op[0]  = c0[v];
    op[16] = c1[v];
    op[32] = c2[v];
    op[48] = c3[v];
  }
}

// ---------------------------------------------------------------------------
// 2) alpha_s/alpha_d[n,h] = <h[n,h,:], a_src/dst[h,:]>  (C fixed = 128).
//    One wave per (node, head); float4 loads + wave32 shuffle reduction.
// ---------------------------------------------------------------------------
__global__ __launch_bounds__(128)
void alpha_kernel(const float* __restrict__ Hf, const float* __restrict__ asrc,
                  const float* __restrict__ adst, float* __restrict__ as_out,
                  float* __restrict__ ad_out, int Hh, int M, int total) {
  const int wid  = blockIdx.x * 4 + (threadIdx.x >> 5);
  const int lane = threadIdx.x & 31;
  if (wid >= total) return;
  const int n  = wid / Hh;
  const int hd = wid - n * Hh;

  const float4 hv = *(const float4*)(Hf + (size_t)n * M + hd * 128 + lane * 4);
  const float4 sv = *(const float4*)(asrc + hd * 128 + lane * 4);
  const float4 dv = *(const float4*)(adst + hd * 128 + lane * 4);
  float s = hv.x * sv.x + hv.y * sv.y + hv.z * sv.z + hv.w * sv.w;
  float d = hv.x * dv.x + hv.y * dv.y + hv.z * dv.z + hv.w * dv.w;
#pragma unroll
  for (int off = 16; off > 0; off >>= 1) {
    s += __shfl_xor(s, off);
    d += __shfl_xor(d, off);
  }
  if (lane == 0) { as_out[wid] = s; ad_out[wid] = d; }
}

// ---------------------------------------------------------------------------
// 3) Segment max of leaky-relu logits (ordered-uint atomicMax).
//    Edges e >= E0 are the implicit self-loops (src=dst=e-E0).
// ---------------------------------------------------------------------------
__global__ void edge_max_kernel(const float* __restrict__ as,
                                const float* __restrict__ ad,
                                const int* __restrict__ esrc,
                                const int* __restrict__ edst,
                                unsigned* __restrict__ mlog,
                                int Hh, int E0, int totalEH) {
  int i = blockIdx.x * blockDim.x + threadIdx.x;
  if (i >= totalEH) return;
  const int e  = i / Hh;
  const int hd = i - e * Hh;
  int s, d;
  if (e < E0) { s = esrc[e]; d = edst[e]; } else { s = d = e - E0; }
  float l = as[s * Hh + hd] + ad[d * Hh + hd];
  l = (l > 0.0f) ? l : NEG_SLOPE * l;
  atomicMax(&mlog[d * Hh + hd], enc_ord(l));
}

__global__ void decode_max_kernel(const unsigned* __restrict__ mlog,
                                  float* __restrict__ mfl, int nh) {
  int i = blockIdx.x * blockDim.x + threadIdx.x;
  if (i >= nh) return;
  float m = dec_ord(mlog[i]);
  mfl[i] = isfinite(m) ? m : 0.0f;   // mirror reference's where(isfinite)
}

// ---------------------------------------------------------------------------
// 4) ex = exp(logit - m[dst]); den[dst] += ex  (ex cached for the aggregate).
// ---------------------------------------------------------------------------
__global__ void edge_exp_kernel(const float* __restrict__ as,
                                const float* __restrict__ ad,
                                const float* __restrict__ mfl,
                                const int* __restrict__ esrc,
                                const int* __restrict__ edst,
                                float* __restrict__ exb,
                                float* __restrict__ den,
                                int Hh, int E0, int totalEH) {
  int i = blockIdx.x * blockDim.x + threadIdx.x;
  if (i >= totalEH) return;
  const int e  = i / Hh;
  const int hd = i - e * Hh;
  int s, d;
  if (e < E0) { s = esrc[e]; d = edst[e]; } else { s = d = e - E0; }
  float l = as[s * Hh + hd] + ad[d * Hh + hd];
  l = (l > 0.0f) ? l : NEG_SLOPE * l;
  float ex = expf(l - mfl[d * Hh + hd]);
  exb[i] = ex;
  atomicAdd(&den[d * Hh + hd], ex);
}

// ---------------------------------------------------------------------------
// 5) acc[dst] += h[src] * alpha.  One wave per edge; coalesced float4 gathers
//    (h[src] row is 2KB), fp32 atomic adds resolve in L2 (acc is L2-resident).
// ---------------------------------------------------------------------------
__global__ __launch_bounds__(128)
void aggregate_kernel(const float* __restrict__ Hf, const float* __restrict__ exb,
                      const float* __restrict__ den,
                      const int* __restrict__ esrc, const int* __restrict__ edst,
                      float* __restrict__ acc, int Hh, int M, int E0, int ET) {
  const int e    = blockIdx.x * 4 + (threadIdx.x >> 5);
  const int lane = threadIdx.x & 31;
  if (e >= ET) return;
  int s, d;
  if (e < E0) { s = esrc[e]; d = edst[e]; } else { s = d = e - E0; }

  for (int j = 0; j < Hh; ++j) {            // one 128-ch head per chunk
    const float a = exb[(size_t)e * Hh + j] / (den[d * Hh + j] + 1e-16f);
    const int c = j * 128 + lane * 4;
    const float4 hv = *(const float4*)(Hf + (size_t)s * M + c);
    float* __restrict__ op = acc + (size_t)d * M + c;
    atomicAdd(op + 0, hv.x * a);
    atomicAdd(op + 1, hv.y * a);
    atomicAdd(op + 2, hv.z * a);
    atomicAdd(op + 3, hv.w * a);
  }
}

// ---------------------------------------------------------------------------
// 6) out = acc + bias  (+ELU for layers 0,1; layer 2 heads=1 mean==identity).
// ---------------------------------------------------------------------------
__global__ void finalize_kernel(const float* __restrict__ acc,
                                const float* __restrict__ bias,
                                float* __restrict__ out, int M, int doElu, int nm) {
  int i = blockIdx.x * blockDim.x + threadIdx.x;
  if (i >= nm) return;
  const int c = i % M;
  float v = acc[i] + bias[c];
  if (doElu) v = (v > 0.0f) ? v : expm1f(v);
  out[i] = v;
}

// ---------------------------------------------------------------------------
// Host-side layer driver (all launches on `stream`, graph-capture safe).
// ---------------------------------------------------------------------------
static void run_layer(hipStream_t stream, const float* xin, const float* W,
                      const float* asrc, const float* adst, const float* bias,
                      int N, int K, int M, int Hh, int E0, int ET,
                      const int* esrc, const int* edst, bool elu,
                      float* hbuf, float* acc, float* asb, float* adb,
                      unsigned* mlog, float* mfl, float* den, float* exb,
                      float* out) {
  const int nm = N * M;
  const int nh = N * Hh;
  const int teh = ET * Hh;

  init_layer_kernel<<<(nm + 255) / 256, 256, 0, stream>>>(acc, nm, den, mlog, nh);

  dim3 ggrid(N / 16, (M + 255) / 256);
  gemm_wmma_f32<<<ggrid, 128, 0, stream>>>(xin, W, hbuf, K, M);

  alpha_kernel<<<(nh + 3) / 4, 128, 0, stream>>>(hbuf, asrc, adst, asb, adb, Hh, M, nh);

  edge_max_kernel<<<(teh + 255) / 256, 256, 0, stream>>>(asb, adb, esrc, edst, mlog,
                                                         Hh, E0, teh);
  decode_max_kernel<<<(nh + 255) / 256, 256, 0, stream>>>(mlog, mfl, nh);

  edge_exp_kernel<<<(teh + 255) / 256, 256, 0, stream>>>(asb, adb, mfl, esrc, edst,
                                                         exb, den, Hh, E0, teh);

  aggregate_kernel<<<(ET + 3) / 4, 128, 0, stream>>>(hbuf, exb, den, esrc, edst,
                                                     acc, Hh, M, E0, ET);

  finalize_kernel<<<(nm + 255) / 256, 256, 0, stream>>>(acc, bias, out, M,
                                                        elu ? 1 : 0, nm);
}

extern "C" void kernel_launch(void* const* d_in, const int* in_sizes, int n_in,
                              void* d_out, int out_size, void* d_ws, size_t ws_size,
                              hipStream_t stream) {
  // Inputs (setup_inputs order)
  const float* x      = (const float*)d_in[0];
  const int*   ei     = (const int*)d_in[1];      // [2, E0]
  const float* W0     = (const float*)d_in[2];
  const float* a_src0 = (const float*)d_in[3];
  const float* a_dst0 = (const float*)d_in[4];
  const float* b0     = (const float*)d_in[5];
  const float* W1     = (const float*)d_in[6];
  const float* a_src1 = (const float*)d_in[7];
  const float* a_dst1 = (const float*)d_in[8];
  const float* b1     = (const float*)d_in[9];
  const float* W2     = (const float*)d_in[10];
  const float* a_src2 = (const float*)d_in[11];
  const float* a_dst2 = (const float*)d_in[12];
  const float* b2     = (const float*)d_in[13];

  const int N  = in_sizes[0] / 256;   // 10000
  const int E0 = in_sizes[1] / 2;     // 320000
  const int ET = E0 + N;              // + self loops
  const int* esrc = ei;
  const int* edst = ei + E0;

  // Workspace carve-up (fp32 words)
  float* ws = (float*)d_ws;
  size_t p = 0;
  float*    hbuf = ws + p;            p += (size_t)N * 512;
  float*    acc  = ws + p;            p += (size_t)N * 512;
  float*    xbuf = ws + p;            p += (size_t)N * 512;
  float*    asb  = ws + p;            p += (size_t)N * 4;
  float*    adb  = ws + p;            p += (size_t)N * 4;
  unsigned* mlog = (unsigned*)(ws + p); p += (size_t)N * 4;
  float*    mfl  = ws + p;            p += (size_t)N * 4;
  float*    den  = ws + p;            p += (size_t)N * 4;
  float*    exb  = ws + p;            p += (size_t)ET * 4;
  (void)ws_size; (void)n_in; (void)out_size;

  // Layer 0: 256 -> 4x128 concat, ELU
  run_layer(stream, x, W0, a_src0, a_dst0, b0, N, 256, 512, 4, E0, ET,
            esrc, edst, true, hbuf, acc, asb, adb, mlog, mfl, den, exb, xbuf);
  // Layer 1: 512 -> 4x128 concat, ELU (xbuf read by GEMM, overwritten by finalize)
  run_layer(stream, xbuf, W1, a_src1, a_dst1, b1, N, 512, 512, 4, E0, ET,
            esrc, edst, true, hbuf, acc, asb, adb, mlog, mfl, den, exb, xbuf);
  // Layer 2: 512 -> 128, heads=1, no concat (mean over 1 head == identity)
  run_layer(stream, xbuf, W2, a_src2, a_dst2, b2, N, 512, 128, 1, E0, ET,
            esrc, edst, false, hbuf, acc, asb, adb, mlog, mfl, den, exb,
            (float*)d_out);
}